// OutlookAttention_47244640256388
// MI455X (gfx1250) — compile-verified
//
#include <hip/hip_runtime.h>
#include <hip/hip_bf16.h>

// ---------------------------------------------------------------------------
// OutlookAttention (VOLO) forward for MI455X (gfx1250, wave32, WMMA).
// GEMMs: V_WMMA_F32_16X16X4_F32, B-operand staged to LDS via async-to-LDS
// (ASYNCcnt double buffering); B fragments read with ds_load_b64 (explicit
// AS3 pointers), A loads with immediate-offset global_load_b64.
// ---------------------------------------------------------------------------

typedef __attribute__((ext_vector_type(2))) float v2f;
typedef __attribute__((ext_vector_type(8))) float v8f;

// explicit LDS (AS3) types so B reads lower to ds_load, not flat_load
typedef __attribute__((address_space(3))) float  sfloat;
typedef __attribute__((address_space(3))) v2f    sv2f;

#define ATT_SCALE 0.17677669529663687f   // 32^-0.5

#if __has_builtin(__builtin_amdgcn_global_load_async_to_lds_b128) && \
    __has_builtin(__builtin_amdgcn_s_wait_asynccnt)
#define USE_ASYNC_LDS 1
#else
#define USE_ASYNC_LDS 0
#endif

// async-to-LDS builtin takes: (v4i global*, v4i lds*, imm offset, imm cpol)
typedef int vi4 __attribute__((vector_size(16)));
typedef __attribute__((address_space(1))) vi4 gvi4;
typedef __attribute__((address_space(3))) vi4 lvi4;

__device__ __forceinline__ void copy16_to_lds(const float* g, sfloat* l)
{
#if USE_ASYNC_LDS
    __builtin_amdgcn_global_load_async_to_lds_b128((gvi4*)g, (lvi4*)l, 0, 0);
#else
    *(lvi4*)l = *(const vi4*)g;      // plain builtin-vector copy (host-safe)
#endif
}

__device__ __forceinline__ void wait_async_all()
{
#if USE_ASYNC_LDS
    __builtin_amdgcn_s_wait_asynccnt(0);
#endif
}

// ---------------------------------------------------------------------------
// WMMA GEMM:  D[M,N] = A[M,K] * Bw[N,K]^T (+ bias[N])
//  - block = 256 threads = 8 waves; wave w owns M-tile blockIdx.x*8+w
//  - all 8 waves share one N-group of NT 16-wide tiles (blockIdx.y)
//  - B chunk [16*NT rows][64 K] staged to LDS (stride 68: bank-conflict-free),
//    double buffered, async copies overlapped with the 16 WMMA steps/chunk.
//  Requires: M % 128 == 0 handled by grid, N % (16*NT) == 0, K % 64 == 0.
// ---------------------------------------------------------------------------
template<int NT, int Nc, int Kc>
__global__ __launch_bounds__(256)
void wmma_gemm_lds(const float* __restrict__ A,
                   const float* __restrict__ Bw,
                   const float* __restrict__ bias,
                   float* __restrict__ D)
{
    constexpr int KC    = 64;            // K-chunk
    constexpr int LSTR  = KC + 4;        // padded LDS row stride (floats)
    constexpr int ROWS  = 16 * NT;       // B rows per N-group
    constexpr int NCH   = Kc / KC;       // chunks

    extern __shared__ float smem[];
    sfloat* sbase = (sfloat*)smem;       // AS3 view of dynamic LDS
    sfloat* buf0  = sbase;
    sfloat* buf1  = sbase + ROWS * LSTR;

    const int tid   = threadIdx.x;
    const int lane  = tid & 31;
    const int wv    = tid >> 5;
    const int m0    = (blockIdx.x * 8 + wv) * 16;
    const int n0    = blockIdx.y * ROWS;
    const int nlane = lane & 15;
    const int khalf = (lane >> 4) * 2;   // lanes 0-15: K+0..1, 16-31: K+2..3
    const int mrow  = m0 + nlane;

    v8f zero = {};
    v8f acc[NT];
#pragma unroll
    for (int j = 0; j < NT; ++j) acc[j] = zero;

    const float* aptr = A + (size_t)mrow * Kc + khalf;
    const float* bsrc = Bw + (size_t)n0 * Kc;

    // cooperative stage of one K-chunk of B into LDS (exactly NT iters/thread)
    auto stage = [&](sfloat* dst, int ch) {
#pragma unroll
        for (int i = 0; i < NT; ++i) {
            const int i4  = tid + i * 256;
            const int row = i4 >> 4;              // / (KC/4)
            const int c4  = i4 & 15;
            copy16_to_lds(bsrc + (size_t)row * Kc + ch * KC + c4 * 4,
                          dst + row * LSTR + c4 * 4);
        }
    };

    auto compute = [&](const sfloat* bufc, const float* ap) {
        v2f afr[KC / 4];
#pragma unroll
        for (int s = 0; s < KC / 4; ++s)          // clause of A loads (imm offs)
            afr[s] = *(const v2f*)(ap + s * 4);
        const sfloat* bb = bufc + nlane * LSTR + khalf;
#pragma unroll
        for (int s = 0; s < KC / 4; ++s) {
            v2f bf[NT];
#pragma unroll
            for (int j = 0; j < NT; ++j)          // ds_load_b64, imm offsets
                bf[j] = *(const sv2f*)(bb + j * 16 * LSTR + s * 4);
#pragma unroll
            for (int j = 0; j < NT; ++j)
                acc[j] = __builtin_amdgcn_wmma_f32_16x16x4_f32(
                    false, afr[s], false, bf[j], (short)0, acc[j], false, false);
        }
    };

    stage(buf0, 0);
    wait_async_all();
    __syncthreads();

    sfloat* bufs[2] = { buf0, buf1 };
    for (int ch = 0; ch < NCH; ++ch) {
        if (ch + 1 < NCH) stage(bufs[(ch + 1) & 1], ch + 1);   // overlap
        compute(bufs[ch & 1], aptr + ch * KC);
        if (ch + 1 < NCH) wait_async_all();
        __syncthreads();
    }

    // D layout: lane holds D[rbase+jv][nlane], rbase = 8*(lane>>4)
    const int rbase = (lane >> 4) * 8;
#pragma unroll
    for (int j = 0; j < NT; ++j) {
        const int ncol = n0 + j * 16 + nlane;
        const float bv = bias ? bias[ncol] : 0.0f;
#pragma unroll
        for (int jv = 0; jv < 8; ++jv)
            D[(size_t)(m0 + rbase + jv) * Nc + ncol] = acc[j][jv] + bv;
    }
}

// ---------------------------------------------------------------------------
// pooled[n,c] = mean over x rows 2n..2n+1, all 300 cols    (N=128, C=384)
// ---------------------------------------------------------------------------
__global__ __launch_bounds__(384)
void pool_kernel(const float* __restrict__ x, float* __restrict__ pooled)
{
    const int n = blockIdx.x;
    const int c = threadIdx.x;
    const float* p = x + (size_t)(2 * n) * 300 * 384 + c;
    float s = 0.0f;
    for (int i = 0; i < 600; ++i)
        s += p[(size_t)i * 384];
    pooled[n * 384 + c] = s * (1.0f / 600.0f);
}

// ---------------------------------------------------------------------------
// In-place scaled softmax over 4608 rows of 900 (rows contiguous in a)
// ---------------------------------------------------------------------------
__global__ __launch_bounds__(256)
void softmax_kernel(float* __restrict__ a)
{
    float* p = a + (size_t)blockIdx.x * 900;
    const int t = threadIdx.x;
    __shared__ float red[256];

    float lmax = -1e30f;
    for (int i = t; i < 900; i += 256)
        lmax = fmaxf(lmax, p[i] * ATT_SCALE);
    red[t] = lmax; __syncthreads();
    for (int s = 128; s > 0; s >>= 1) {
        if (t < s) red[t] = fmaxf(red[t], red[t + s]);
        __syncthreads();
    }
    const float m = red[0];
    __syncthreads();

    float lsum = 0.0f;
    for (int i = t; i < 900; i += 256) {
        float e = __expf(p[i] * ATT_SCALE - m);
        p[i] = e;
        lsum += e;
    }
    red[t] = lsum; __syncthreads();
    for (int s = 128; s > 0; s >>= 1) {
        if (t < s) red[t] += red[t + s];
        __syncthreads();
    }
    const float inv = 1.0f / red[0];
    for (int i = t; i < 900; i += 256) p[i] *= inv;
}

// ---------------------------------------------------------------------------
// y[h,n,k0,d] = sum_m a[(n,h,k0),m] * v_win(h,n,m,d)
//   m = k0v*300 + k1v ; v_win = v[row 2n+k0v-1, col k1v-1, ch h*32+d] (0 pad)
// ---------------------------------------------------------------------------
__global__ __launch_bounds__(128)
void y_kernel(const float* __restrict__ a, const float* __restrict__ v,
              float* __restrict__ y)
{
    const int h = blockIdx.x;    // 12
    const int n = blockIdx.y;    // 128
    const int t = threadIdx.x;
    if (t >= 96) return;
    const int k0 = t >> 5;
    const int d  = t & 31;

    const float* arow = a + (size_t)n * 32400 + (h * 3 + k0) * 900;
    float s = 0.0f;
#pragma unroll
    for (int k0v = 0; k0v < 3; ++k0v) {
        const int r = 2 * n + k0v;              // padded row
        if (r >= 1 && r <= 256) {
            const float* am   = arow + k0v * 300;
            const float* vrow = v + (size_t)(r - 1) * 300 * 384 + h * 32 + d;
            for (int k1v = 1; k1v < 300; ++k1v)     // k1v==0 is pad column
                s += am[k1v] * vrow[(size_t)(k1v - 1) * 384];
        }
    }
    y[(((size_t)h * 128 + n) * 3 + k0) * 32 + d] = s;
}

// ---------------------------------------------------------------------------
// Fold (col2im with the reference's tile/reshape scramble):
// outp[hout, w, c=h*32+c_off] = sum_{k0p: r-k0p even, 0<=n<128}
//      y[h, (q%96)/3, q%3, n],  q = c_off*900 + k0p*300 + (w+1), r = hout+1
// ---------------------------------------------------------------------------
__global__ __launch_bounds__(384)
void fold_kernel(const float* __restrict__ y, float* __restrict__ outp)
{
    const int w    = blockIdx.x;   // 0..299
    const int hout = blockIdx.y;   // 0..255
    const int c    = threadIdx.x;  // 0..383
    const int h    = c >> 5;
    const int coff = c & 31;

    float s = 0.0f;
    if (w < 299) {
        const int k1 = w + 1;
        const int r  = hout + 1;
#pragma unroll
        for (int k0p = 0; k0p < 3; ++k0p) {
            const int num = r - k0p;
            if (num >= 0 && (num & 1) == 0) {
                const int n = num >> 1;
                if (n < 128) {
                    const int q   = coff * 900 + k0p * 300 + k1;
                    const int q96 = q % 96;
                    const int dq  = q96 / 3;
                    const int kq  = q96 - dq * 3;
                    s += y[(((size_t)h * 128 + n) * 3 + kq) * 32 + dq];
                }
            }
        }
    }
    outp[((size_t)hout * 300 + w) * 384 + c] = s;
}

// ---------------------------------------------------------------------------
extern "C" void kernel_launch(void* const* d_in, const int* in_sizes, int n_in,
                              void* d_out, int out_size, void* d_ws, size_t ws_size,
                              hipStream_t stream)
{
    const float* x      = (const float*)d_in[0];   // [256*300, 384]
    const float* w_v    = (const float*)d_in[1];   // [384, 384]
    const float* w_attn = (const float*)d_in[2];   // [32400, 384]
    const float* b_attn = (const float*)d_in[3];   // [32400]
    const float* w_proj = (const float*)d_in[4];   // [384, 384]
    const float* b_proj = (const float*)d_in[5];   // [384]
    float* out = (float*)d_out;                    // [76800, 384]

    const int HW = 256 * 300;                      // 76800
    float* ws = (float*)d_ws;
    float* v      = ws;                            // 29,491,200 f
    float* foldbf = v + (size_t)HW * 384;          // 29,491,200 f
    float* a      = foldbf + (size_t)HW * 384;     //  4,147,200 f
    float* pooled = a + (size_t)128 * 32400;       //     49,152 f
    float* y      = pooled + 128 * 384;            //    147,456 f

    const size_t lds12 = 2u * (16 * 12) * 68 * sizeof(float); // 104,448 B
    const size_t lds9  = 2u * (16 * 9)  * 68 * sizeof(float); //  78,336 B

    // 1) v = x * w_v^T   (M=76800, N=384, K=384; NT=12 -> 2 N-groups)
    wmma_gemm_lds<12, 384, 384><<<dim3(HW / 128, 2), 256, lds12, stream>>>(
        x, w_v, nullptr, v);

    // 2) 2x300 average pool
    pool_kernel<<<128, 384, 0, stream>>>(x, pooled);

    // 3) a = pooled * w_attn^T + b_attn (M=128, N=32400; NT=9 -> 225 groups)
    wmma_gemm_lds<9, 32400, 384><<<dim3(1, 225), 256, lds9, stream>>>(
        pooled, w_attn, b_attn, a);

    // 4) softmax over 4608 rows of 900 (in place)
    softmax_kernel<<<4608, 256, 0, stream>>>(a);

    // 5) y = a * v_win
    y_kernel<<<dim3(12, 128), 128, 0, stream>>>(a, v, y);

    // 6) fold/scatter into [HW, 384]
    fold_kernel<<<dim3(300, 256), 384, 0, stream>>>(y, foldbf);

    // 7) out = fold * w_proj^T + b_proj
    wmma_gemm_lds<12, 384, 384><<<dim3(HW / 128, 2), 256, lds12, stream>>>(
        foldbf, w_proj, b_proj, out);

    (void)in_sizes; (void)n_in; (void)out_size; (void)ws_size;
}